// DualQuantizer_53266184405829
// MI455X (gfx1250) — compile-verified
//
#include <hip/hip_runtime.h>
#include <float.h>

// Problem dims
#define B_DIM   8
#define T_DIM   4096
#define D_SEM   256
#define D_AC    36
#define D_ALL   292
#define N_CB    8192
#define M_ROWS  (B_DIM * T_DIM)          // 32768

// d_out layout (floats, concat in return order)
#define OFF_ZQ     0
#define OFF_IDX    (B_DIM * D_ALL * T_DIM)            // 9568256
#define OFF_CODES  (OFF_IDX + M_ROWS)                 // 9601024
#define OFF_LOSS   (OFF_CODES + B_DIM * D_AC * T_DIM) // 10780672
#define OFF_ZACQ   (OFF_LOSS + 1)

// d_ws layout (float units)
#define WS_C2    0            // 8192 floats: per-codebook-row squared norm
#define WS_PART  8192         // 2048 floats: per-block loss partials
#define WS_IDX   10240        // 32768 ints: argmin indices

// LDS tile: K-pair-major transposed codebook tile.
// lds[k2 * ROWLEN + 2*n + (k&1)], k2 = k/2 in 0..127, n in 0..31.
// ROWLEN=66 skews banks: staging writes hit bank (l + 2n) -> conflict-free.
#define ROWLEN  66

typedef __attribute__((ext_vector_type(2))) float v2f;
typedef __attribute__((ext_vector_type(8))) float v8f;

// ---------------------------------------------------------------------------
// Kernel 1: c2[n] = sum_k codebook[n][k]^2  (wave-per-row, coalesced)
// ---------------------------------------------------------------------------
__global__ void prep_c2_kernel(const float* __restrict__ cb, float* __restrict__ ws) {
  const int lane = threadIdx.x & 31;
  const int wave = threadIdx.x >> 5;
  const int n = blockIdx.x * 8 + wave;
  const float* row = cb + n * D_SEM;
  float s = 0.f;
#pragma unroll
  for (int j = 0; j < D_SEM / 32; ++j) {
    float v = row[lane + 32 * j];
    s += v * v;
  }
#pragma unroll
  for (int off = 16; off >= 1; off >>= 1) s += __shfl_xor(s, off, 32);
  if (lane == 0) ws[WS_C2 + n] = s;
}

// ---------------------------------------------------------------------------
// Kernel 2: distance GEMM via V_WMMA_F32_16X16X4_F32 + fused argmin.
// Block = 256 threads (8 waves), 128 rows of z per block; wave owns 16 rows.
// A panel (16x256 f32) lives in registers; codebook staged pair-major in LDS
// so each B fragment is ONE ds_load_b64 straight into an even VGPR pair.
// ---------------------------------------------------------------------------
__global__ void __launch_bounds__(256, 1)
vq_argmin_kernel(const float* __restrict__ z, const float* __restrict__ cb,
                 const float* __restrict__ c2, int* __restrict__ ws_idx,
                 float* __restrict__ out_idx_f) {
  __shared__ float lds_cb[128 * ROWLEN];   // 33792 B

  const int lane = threadIdx.x & 31;
  const int wave = threadIdx.x >> 5;
  const int M0 = blockIdx.x * 128 + wave * 16;

  // --- preload A fragments: 64 x v2f covering 16 rows x K=256 ---
  // A layout (32-bit 16x4): lanes 0-15 / 16-31 both hold M=0..15;
  // VGPR0 = K=k0 (lo lanes) / K=k0+2 (hi lanes); VGPR1 = K=k0+1 / k0+3.
  const int row = M0 + (lane & 15);
  const float* zrow = z + (row >> 12) * (D_ALL * T_DIM) + (row & 4095);
  const int kb = (lane >> 4) << 1;       // 0 for lanes 0-15, 2 for lanes 16-31
  v2f a[64];
#pragma unroll
  for (int kk = 0; kk < 64; ++kk) {
    a[kk].x = zrow[(4 * kk + kb) * T_DIM];
    a[kk].y = zrow[(4 * kk + kb + 1) * T_DIM];
  }

  float best[8];
  int bidx[8];
#pragma unroll
  for (int r = 0; r < 8; ++r) { best[r] = FLT_MAX; bidx[r] = 0x7fffffff; }

  // Per-lane base pointers into the LDS tile for the two N-subtiles:
  // fragment kk lives at v2f offset kk*ROWLEN from this base.
  const v2f* bp0 = (const v2f*)&lds_cb[(lane >> 4) * ROWLEN + 2 * (lane & 15)];
  const v2f* bp1 = (const v2f*)&lds_cb[(lane >> 4) * ROWLEN + 32 + 2 * (lane & 15)];

  for (int n0 = 0; n0 < N_CB; n0 += 32) {
    __syncthreads();
    // stage 32 codebook rows, transposed & K-pair-packed:
    // lds[(k>>1)*ROWLEN + 2*n + (k&1)] = cb[n0+n][k]
    for (int i = threadIdx.x; i < 32 * D_SEM; i += 256) {
      int n = i >> 8;
      int k = i & 255;
      lds_cb[(k >> 1) * ROWLEN + 2 * n + (k & 1)] = cb[(n0 + n) * D_SEM + k];
    }
    __syncthreads();

    // fire-and-forget prefetch of next tile (32 rows = 32 KB, 128 B/thread)
    if (n0 + 32 < N_CB) {
      __builtin_prefetch(&cb[(n0 + 32) * D_SEM + threadIdx.x * 32], 0, 0);
    }

#pragma unroll
    for (int sub = 0; sub < 2; ++sub) {
      const int n = n0 + sub * 16 + (lane & 15);
      const float cterm = c2[n];
      const v2f* bp = sub ? bp1 : bp0;
      v8f acc = {0.f, 0.f, 0.f, 0.f, 0.f, 0.f, 0.f, 0.f};
#pragma unroll
      for (int kk = 0; kk < 64; ++kk) {
        v2f bf = bp[kk * ROWLEN];     // single ds_load_b64: {K, K+1} for this lane
        acc = __builtin_amdgcn_wmma_f32_16x16x4_f32(
            false, a[kk], false, bf, (short)0, acc, false, false);
      }
      // C layout: VGPR r holds M=r (lanes 0-15) / M=r+8 (lanes 16-31), N=lane&15
#pragma unroll
      for (int r = 0; r < 8; ++r) {
        float s = fmaf(-2.0f, acc[r], cterm);
        if (s < best[r] || (s == best[r] && n < bidx[r])) {
          best[r] = s;
          bidx[r] = n;
        }
      }
    }
  }

  // argmin reduction across the 16 lanes of each half-wave
#pragma unroll
  for (int off = 8; off >= 1; off >>= 1) {
#pragma unroll
    for (int r = 0; r < 8; ++r) {
      float ob = __shfl_xor(best[r], off, 16);
      int oi = __shfl_xor(bidx[r], off, 16);
      if (ob < best[r] || (ob == best[r] && oi < bidx[r])) {
        best[r] = ob;
        bidx[r] = oi;
      }
    }
  }

  if ((lane & 15) == 0) {
    const int mbase = M0 + ((lane >> 4) << 3);  // lane0 -> M0..M0+7, lane16 -> M0+8..+15
#pragma unroll
    for (int r = 0; r < 8; ++r) {
      ws_idx[mbase + r] = bidx[r];
      out_idx_f[mbase + r] = (float)bidx[r];
    }
  }
}

// ---------------------------------------------------------------------------
// Kernel 3: z_q gather + deterministic per-block loss partials
// ---------------------------------------------------------------------------
__global__ void gather_loss_kernel(const float* __restrict__ z,
                                   const float* __restrict__ cb,
                                   const int* __restrict__ idx,
                                   float* __restrict__ out,
                                   float* __restrict__ partials) {
  const int total = B_DIM * D_SEM * T_DIM;             // 8388608
  const int stride = gridDim.x * blockDim.x;           // 524288 -> 16 iters
  float acc = 0.f;
  for (int e = blockIdx.x * blockDim.x + threadIdx.x; e < total; e += stride) {
    int b = e >> 20;                                   // / (256*4096)
    int rem = e & ((1 << 20) - 1);
    int d = rem >> 12;
    int t = rem & 4095;
    int id = idx[(b << 12) + t];
    float q = cb[id * D_SEM + d];
    int zo = b * (D_ALL * T_DIM) + d * T_DIM + t;
    float diff = q - z[zo];
    acc += diff * diff;
    out[OFF_ZQ + zo] = q;                              // z_q_ste forward value == z_q
  }
  // deterministic block reduction
  __shared__ float sm[8];
#pragma unroll
  for (int off = 16; off >= 1; off >>= 1) acc += __shfl_xor(acc, off, 32);
  const int lane = threadIdx.x & 31;
  const int wave = threadIdx.x >> 5;
  if (lane == 0) sm[wave] = acc;
  __syncthreads();
  if (wave == 0) {
    float v = (lane < 8) ? sm[lane] : 0.f;
#pragma unroll
    for (int off = 4; off >= 1; off >>= 1) v += __shfl_xor(v, off, 8);
    if (lane == 0) partials[blockIdx.x] = v;
  }
}

// ---------------------------------------------------------------------------
// Kernel 4: FSQ on the 36 acoustic channels
// ---------------------------------------------------------------------------
__global__ void fsq_kernel(const float* __restrict__ z, float* __restrict__ out) {
  const int total = B_DIM * D_AC * T_DIM;              // 1179648
  int e = blockIdx.x * blockDim.x + threadIdx.x;
  if (e >= total) return;
  int b = e / (D_AC * T_DIM);
  int rem = e - b * (D_AC * T_DIM);
  int d = rem >> 12;
  int t = rem & 4095;
  float zac = z[b * (D_ALL * T_DIM) + (D_SEM + d) * T_DIM + t];
  float zb = tanhf(zac) * 10.001f;                     // (HALF + EPS)
  float zq = rintf(zb);                                // round half to even == jnp.round
  int code = (int)(zq + 10.0f);
  code = code < 0 ? 0 : (code > 20 ? 20 : code);
  out[OFF_ZQ + b * (D_ALL * T_DIM) + (D_SEM + d) * T_DIM + t] = zq;
  out[OFF_CODES + e] = (float)code;
  out[OFF_ZACQ + e] = zq;
}

// ---------------------------------------------------------------------------
// Kernel 5: fixed-order final loss reduction (deterministic across replays)
// ---------------------------------------------------------------------------
__global__ void finalize_kernel(const float* __restrict__ partials,
                                float* __restrict__ out) {
  float s = 0.f;
#pragma unroll
  for (int j = 0; j < 8; ++j) s += partials[threadIdx.x + 256 * j];  // 2048 partials
  __shared__ float sm[8];
#pragma unroll
  for (int off = 16; off >= 1; off >>= 1) s += __shfl_xor(s, off, 32);
  const int lane = threadIdx.x & 31;
  const int wave = threadIdx.x >> 5;
  if (lane == 0) sm[wave] = s;
  __syncthreads();
  if (threadIdx.x == 0) {
    float tot = 0.f;
#pragma unroll
    for (int j = 0; j < 8; ++j) tot += sm[j];
    // loss = codebook_loss + 0.25 * commitment_loss, identical MSE values
    out[OFF_LOSS] = 1.25f * tot / (float)(B_DIM * D_SEM * T_DIM);
  }
}

// ---------------------------------------------------------------------------
extern "C" void kernel_launch(void* const* d_in, const int* in_sizes, int n_in,
                              void* d_out, int out_size, void* d_ws, size_t ws_size,
                              hipStream_t stream) {
  const float* z  = (const float*)d_in[0];   // (8, 292, 4096) f32
  const float* cb = (const float*)d_in[1];   // (8192, 256) f32
  float* out = (float*)d_out;
  float* wsf = (float*)d_ws;
  float* c2       = wsf + WS_C2;
  float* partials = wsf + WS_PART;
  int*   idx      = (int*)(wsf + WS_IDX);

  prep_c2_kernel<<<N_CB / 8, 256, 0, stream>>>(cb, wsf);
  vq_argmin_kernel<<<M_ROWS / 128, 256, 0, stream>>>(z, cb, c2, idx, out + OFF_IDX);
  gather_loss_kernel<<<2048, 256, 0, stream>>>(z, cb, idx, out, partials);
  fsq_kernel<<<(B_DIM * D_AC * T_DIM + 255) / 256, 256, 0, stream>>>(z, out);
  finalize_kernel<<<1, 256, 0, stream>>>(partials, out);
}